// CTCBeamSearchDecoder_47132971106740
// MI455X (gfx1250) — compile-verified
//
#include <hip/hip_runtime.h>
#include <stdint.h>

#define B_    8
#define T_    2048
#define V_    4096
#define BEAM_ 8
#define TPB1  256
#define NS    32    // ring slots (steps)
#define DPF   6     // prefetch depth in groups of 4 steps

#define AS1 __attribute__((address_space(1)))
#define AS3 __attribute__((address_space(3)))

typedef int   v2i __attribute__((ext_vector_type(2)));
typedef float v4f __attribute__((ext_vector_type(4)));

// ---- ordered-float encoding: monotone unsigned key for f32 ----
__device__ __forceinline__ unsigned ordf(float f) {
  unsigned u = __float_as_uint(f);
  return (u & 0x80000000u) ? ~u : (u | 0x80000000u);
}
__device__ __forceinline__ float iordf(unsigned o) {
  unsigned u = (o & 0x80000000u) ? (o ^ 0x80000000u) : ~o;
  return __uint_as_float(u);
}

// ---- CDNA5 async global->LDS copy (ASYNCcnt-tracked) ----
__device__ __forceinline__ void async_b64(const void* g, void* l) {
#if __has_builtin(__builtin_amdgcn_global_load_async_to_lds_b64)
  __builtin_amdgcn_global_load_async_to_lds_b64((AS1 v2i*)g, (AS3 v2i*)l, 0, 0);
#else
  unsigned loff = (unsigned)(uintptr_t)(AS3 void*)l;
  asm volatile("global_load_async_to_lds_b64 %0, %1, off"
               :: "v"(loff), "v"(g) : "memory");
#endif
}

#if __has_builtin(__builtin_amdgcn_s_wait_asynccnt)
#define WAIT_ASYNCCNT(n) do { __builtin_amdgcn_s_wait_asynccnt(n); \
                              asm volatile("" ::: "memory"); } while (0)
#else
#define WAIT_ASYNCCNT(n) asm volatile("s_wait_asynccnt %0" :: "i"(n) : "memory")
#endif

// =====================================================================
// Kernel 1: fused log_softmax + per-row top-8 (value, token)
// One block per (b,t) row of V=4096. 256 threads x 16 f32 in registers.
// Streaming data uses TH=NT (non-temporal) so the 2x268MB streams don't
// thrash the 192MB L2; the 1MB top-8 workspace stays RT/L2-resident.
// =====================================================================
__global__ __launch_bounds__(TPB1)
void k_logsoftmax_top8(const float* __restrict__ logits,
                       float* __restrict__ logP,
                       unsigned long long* __restrict__ wsrows) {
  const int row = blockIdx.x;
  const int tid = threadIdx.x;
  const v4f* src4 = reinterpret_cast<const v4f*>(logits + (size_t)row * V_);
  v4f* dst4 = reinterpret_cast<v4f*>(logP + (size_t)row * V_);

  v4f v[4];
#pragma unroll
  for (int k = 0; k < 4; ++k)
    v[k] = __builtin_nontemporal_load(src4 + k * TPB1 + tid);

  __shared__ float redf[8];
  __shared__ unsigned long long redk[8];

  // ---- block max (wave32 shfl + 8-slot LDS) ----
  float m = -INFINITY;
#pragma unroll
  for (int k = 0; k < 4; ++k)
    m = fmaxf(m, fmaxf(fmaxf(v[k].x, v[k].y), fmaxf(v[k].z, v[k].w)));
#pragma unroll
  for (int o = 16; o > 0; o >>= 1) m = fmaxf(m, __shfl_xor(m, o, 32));
  if ((tid & 31) == 0) redf[tid >> 5] = m;
  __syncthreads();
  float mx = redf[0];
#pragma unroll
  for (int w = 1; w < 8; ++w) mx = fmaxf(mx, redf[w]);
  __syncthreads();

  // ---- block sum of exp ----
  float s = 0.f;
#pragma unroll
  for (int k = 0; k < 4; ++k)
    s += __expf(v[k].x - mx) + __expf(v[k].y - mx) +
         __expf(v[k].z - mx) + __expf(v[k].w - mx);
#pragma unroll
  for (int o = 16; o > 0; o >>= 1) s += __shfl_xor(s, o, 32);
  if ((tid & 31) == 0) redf[tid >> 5] = s;
  __syncthreads();
  float tot = redf[0];
#pragma unroll
  for (int w = 1; w < 8; ++w) tot += redf[w];
  __syncthreads();

  const float logZ = mx + logf(tot);

  // ---- write logP (NT), build 64-bit sort keys (value desc, token asc) ----
  unsigned long long key[16];
#pragma unroll
  for (int k = 0; k < 4; ++k) {
    v4f o4;
    o4.x = v[k].x - logZ; o4.y = v[k].y - logZ;
    o4.z = v[k].z - logZ; o4.w = v[k].w - logZ;
    __builtin_nontemporal_store(o4, dst4 + k * TPB1 + tid);
    const unsigned bt = (unsigned)((k * TPB1 + tid) * 4);
    key[k * 4 + 0] = ((unsigned long long)ordf(o4.x) << 12) | (4095u - (bt + 0));
    key[k * 4 + 1] = ((unsigned long long)ordf(o4.y) << 12) | (4095u - (bt + 1));
    key[k * 4 + 2] = ((unsigned long long)ordf(o4.z) << 12) | (4095u - (bt + 2));
    key[k * 4 + 3] = ((unsigned long long)ordf(o4.w) << 12) | (4095u - (bt + 3));
  }

  unsigned long long* wrow = wsrows + (size_t)row * 8;  // 64B row: 8 f32 + 8 i32
#pragma unroll 1
  for (int r = 0; r < 8; ++r) {
    unsigned long long kb = key[0];
#pragma unroll
    for (int i = 1; i < 16; ++i) if (key[i] > kb) kb = key[i];
#pragma unroll
    for (int o = 16; o > 0; o >>= 1) {
      unsigned long long x = __shfl_xor(kb, o, 32);
      if (x > kb) kb = x;
    }
    if ((tid & 31) == 0) redk[tid >> 5] = kb;
    __syncthreads();
    unsigned long long kw = redk[0];
#pragma unroll
    for (int w = 1; w < 8; ++w) if (redk[w] > kw) kw = redk[w];
    __syncthreads();
#pragma unroll
    for (int i = 0; i < 16; ++i) if (key[i] == kw) key[i] = 0ull;
    if (tid == 0) {
      reinterpret_cast<float*>(wrow)[r] = iordf((unsigned)(kw >> 12));
      reinterpret_cast<int*>(wrow)[8 + r] = (int)(4095u - (unsigned)(kw & 0xFFFu));
    }
  }
}

// =====================================================================
// Kernel 2: per-batch beam scan (1 wave), async-LDS prefetch ring,
// tok/bp history in LDS, then backtrace + CTC collapse.
// =====================================================================
__global__ __launch_bounds__(32)
void k_beam_scan(const unsigned long long* __restrict__ wsrows,
                 const int* __restrict__ lengths,
                 float* __restrict__ out_tokens,
                 float* __restrict__ out_lens,
                 float* __restrict__ out_scores) {
  const int b = blockIdx.x;
  const int lane = threadIdx.x;
  const int len = lengths[b];

  __shared__ short tok_s[T_][BEAM_];            // 32 KB
  __shared__ unsigned char bp_s[T_][BEAM_];     // 16 KB
  __shared__ unsigned long long ring[NS][8];    //  2 KB (64B per step)
  __shared__ short path_s[T_];                  //  4 KB

  const unsigned long long* rowbase = wsrows + (size_t)b * T_ * 8;

  // issue one group = 4 steps of 64B rows; 32 lanes x 8B each
  auto issue = [&](int g) {
    const int s = 4 * g + (lane >> 3);
    const int c = lane & 7;
    const int srow = (s < T_) ? s : (T_ - 1);   // clamp tail (garbage slot, never read)
    async_b64((const void*)(rowbase + (size_t)srow * 8 + c),
              (void*)&ring[s & (NS - 1)][c]);
  };
  for (int g = 0; g < DPF; ++g) issue(g);

  float sc[8];
  const int G = T_ / 4;
  for (int g = 0; g < G; ++g) {
    WAIT_ASYNCCNT(DPF - 1);   // in-order completion => group g is resident
    issue(g + DPF);
#pragma unroll 1
    for (int q = 0; q < 4; ++q) {
      const int t = 4 * g + q;
      const float* rv = (const float*)&ring[t & (NS - 1)][0];
      const int* ri = (const int*)&ring[t & (NS - 1)][4];
      float tv[8]; int ti[8];
#pragma unroll
      for (int j = 0; j < 8; ++j) { tv[j] = rv[j]; ti[j] = ri[j]; }

      if (t == 0) {
#pragma unroll
        for (int j = 0; j < 8; ++j) sc[j] = tv[j];
        if (lane < 8) { tok_s[0][lane] = (short)ti[lane]; bp_s[0][lane] = 0; }
      } else if (t < len) {
        // 64 candidates: 2 per lane. key = ord(val)<<15 | (32767 - (beam<<12|tok))
        unsigned long long k0, k1;
        {
          const int c0 = lane, c1 = lane + 32;
          const int b0 = c0 >> 3, j0 = c0 & 7, b1 = c1 >> 3, j1 = c1 & 7;
          const float v0 = sc[b0] + tv[j0], v1 = sc[b1] + tv[j1];
          const int f0 = (b0 << 12) | ti[j0], f1 = (b1 << 12) | ti[j1];
          k0 = ((unsigned long long)ordf(v0) << 15) | (unsigned)(32767 - f0);
          k1 = ((unsigned long long)ordf(v1) << 15) | (unsigned)(32767 - f1);
        }
        float nsc[8];
#pragma unroll
        for (int r = 0; r < 8; ++r) {
          unsigned long long kb = (k0 > k1) ? k0 : k1;
#pragma unroll
          for (int o = 16; o > 0; o >>= 1) {
            unsigned long long x = __shfl_xor(kb, o, 32);
            if (x > kb) kb = x;
          }
          const int flat = 32767 - (int)(kb & 0x7FFFull);
          nsc[r] = iordf((unsigned)(kb >> 15));
          if (lane == 0) {
            tok_s[t][r] = (short)(flat & 4095);
            bp_s[t][r] = (unsigned char)(flat >> 12);
          }
          if (k0 == kb) k0 = 0ull;
          if (k1 == kb) k1 = 0ull;
        }
#pragma unroll
        for (int j = 0; j < 8; ++j) sc[j] = nsc[j];
      } else {
        if (lane < 8) { tok_s[t][lane] = 0; bp_s[t][lane] = (unsigned char)lane; }
      }
    }
  }
  WAIT_ASYNCCNT(0);

  // ---- backtrace best beam (index 0) ----
  if (lane == 0) {
    int beam = 0;
    for (int t = T_ - 1; t >= 0; --t) {
      path_s[t] = tok_s[t][beam];
      beam = bp_s[t][beam];
    }
  }
  // ---- pad with -1, then CTC collapse ----
  for (int i = lane; i < T_; i += 32) out_tokens[(size_t)b * T_ + i] = -1.0f;
  if (lane == 0) {
    int prev = -1, pos = 0;
    for (int t = 0; t < T_; ++t) {
      const int p = path_s[t];
      if (p != 0 && p != prev && t < len) {
        out_tokens[(size_t)b * T_ + pos] = (float)p;
        ++pos;
      }
      prev = p;
    }
    out_lens[b] = (float)pos;
    out_scores[b] = sc[0];
  }
}

// =====================================================================
extern "C" void kernel_launch(void* const* d_in, const int* in_sizes, int n_in,
                              void* d_out, int out_size, void* d_ws, size_t ws_size,
                              hipStream_t stream) {
  (void)in_sizes; (void)n_in; (void)out_size; (void)ws_size;
  const float* logits = (const float*)d_in[0];
  const int* lengths = (const int*)d_in[1];

  float* out = (float*)d_out;
  float* out_tokens = out;                        // [B*T]
  float* out_lens   = out + (size_t)B_ * T_;      // [B]
  float* out_scores = out_lens + B_;              // [B]
  float* logP       = out_scores + B_;            // [B*T*V], offset 16400 (16B aligned)

  unsigned long long* wsrows = (unsigned long long*)d_ws;  // [B*T][8] = 1 MB

  k_logsoftmax_top8<<<B_ * T_, TPB1, 0, stream>>>(logits, logP, wsrows);
  k_beam_scan<<<B_, 32, 0, stream>>>(wsrows, lengths, out_tokens, out_lens, out_scores);
}